// FM_v2_38560216383901
// MI455X (gfx1250) — compile-verified
//
#include <hip/hip_runtime.h>
#include <hip/hip_bf16.h>

typedef float v2f __attribute__((ext_vector_type(2)));
typedef float v8f __attribute__((ext_vector_type(8)));

#define N_COLS 8
#define EMB_NUM 12
#define EMB_DIM 16
#define NPAIR 28
#define BLOCK 256
#define BATCH_PER_WAVE 16
#define BATCH_PER_BLOCK 128

// stable softplus: log1p(exp(x)) = max(x,0) + log(1 + exp(-|x|))
__device__ __forceinline__ float softplus_f(float x) {
    return fmaxf(x, 0.0f) + __logf(1.0f + __expf(-fabsf(x)));
}

// Pass 1: fused gather + reparam + 28-pair FM interaction + WMMA d-reduction.
// Each wave handles 16 batch rows. WMMA chain computes
//   D = G(16x16) * B1 + S(16x16) * B2   via 8x V_WMMA_F32_16X16X4_F32
// where G[b,d] = sum_p w[p,d]*e_i*e_j, S[b,d] = colsum, B1 cols 0/1 = ones,
// B2 cols 0/1 = emb_action rows. D[:,0]=inf0, D[:,1]=inf1.
// f32 A layout (ISA 7.12.2): lanes 0-15 supply K=k0,k0+1; lanes 16-31 K=k0+2,k0+3.
// B assumed lane-symmetric (VGPR0: K=k0 low half / K=k0+2 high half); the
// ones-matrix B1 is layout-invariant by construction.
__global__ __launch_bounds__(BLOCK) void fm_fused_kernel(
    const int*   __restrict__ idx,        // [8, B]
    const float* __restrict__ label,      // [B, 2]
    const float* __restrict__ posw,       // [B, 2]
    const float* __restrict__ noise,      // [B, 16]
    const float* __restrict__ emb_mean,   // [8, 12, 16]
    const float* __restrict__ emb_std,    // [8, 12, 16]
    const float* __restrict__ fc_w,       // [28, 16]
    const float* __restrict__ emb_action, // [2, 16]
    float* __restrict__ out,              // [B, 2]
    float* __restrict__ blk_loss,         // [gridDim.x]
    int Bn)
{
    __shared__ float s_mean [N_COLS * EMB_NUM * EMB_DIM];
    __shared__ float s_std  [N_COLS * EMB_NUM * EMB_DIM];
    __shared__ float s_w    [NPAIR * EMB_DIM];
    __shared__ float s_act  [2 * EMB_DIM];
    __shared__ float s_noise[BATCH_PER_BLOCK * EMB_DIM];
    __shared__ int   s_idx  [N_COLS * BATCH_PER_BLOCK];
    __shared__ float s_red  [BLOCK / 32];

    const int tid    = threadIdx.x;
    const int base_b = blockIdx.x * BATCH_PER_BLOCK;

    // cooperative LDS staging (tables ~14KB + per-block slabs ~12KB)
    for (int q = tid; q < N_COLS * EMB_NUM * EMB_DIM; q += BLOCK) {
        s_mean[q] = emb_mean[q];
        s_std[q]  = emb_std[q];
    }
    for (int q = tid; q < NPAIR * EMB_DIM; q += BLOCK) s_w[q] = fc_w[q];
    if (tid < 2 * EMB_DIM) s_act[tid] = emb_action[tid];
    for (int q = tid; q < BATCH_PER_BLOCK * EMB_DIM; q += BLOCK)
        s_noise[q] = noise[(size_t)base_b * EMB_DIM + q];
    for (int q = tid; q < N_COLS * BATCH_PER_BLOCK; q += BLOCK) {
        int c = q >> 7, t = q & (BATCH_PER_BLOCK - 1);
        s_idx[q] = idx[(size_t)c * Bn + base_b + t];
    }
    __syncthreads();

    const int wave    = tid >> 5;
    const int lane    = tid & 31;
    const int col     = lane & 15;      // N column of WMMA B/D
    const int half    = lane >> 4;      // which K half this lane supplies
    const int koff    = half << 1;      // K offset within a K=4 chunk
    const int b_local = wave * BATCH_PER_WAVE + col;

    int ix[8];
    #pragma unroll
    for (int c = 0; c < 8; ++c) ix[c] = s_idx[c * BATCH_PER_BLOCK + b_local];

    const int PI[NPAIR] = {0,0,0,0,0,0,0,1,1,1,1,1,1,2,2,2,2,2,3,3,3,3,4,4,4,5,5,6};
    const int PJ[NPAIR] = {1,2,3,4,5,6,7,2,3,4,5,6,7,3,4,5,6,7,4,5,6,7,5,6,7,6,7,7};

    // B1: ones in columns 0 and 1 (layout-invariant), zeros elsewhere
    v2f b1; b1.x = (col < 2) ? 1.0f : 0.0f; b1.y = b1.x;
    v8f acc = {0.f, 0.f, 0.f, 0.f, 0.f, 0.f, 0.f, 0.f};

    #pragma unroll
    for (int t4 = 0; t4 < 4; ++t4) {
        const int d0 = t4 * 4 + koff;
        float g[2], cs[2];
        #pragma unroll
        for (int u = 0; u < 2; ++u) {
            const int dd = d0 + u;
            float nv = s_noise[(b_local << 4) + dd] * 0.01f;
            float e[8];
            float c_s = 0.f;
            #pragma unroll
            for (int c = 0; c < 8; ++c) {
                int off = (c * EMB_NUM + ix[c]) * EMB_DIM + dd;
                float ev = fmaf(softplus_f(s_std[off]), nv, s_mean[off]);
                e[c] = ev;
                c_s += ev;
            }
            float gg = 0.f;
            #pragma unroll
            for (int p = 0; p < NPAIR; ++p)
                gg = fmaf(s_w[p * EMB_DIM + dd] * e[PI[p]], e[PJ[p]], gg);
            g[u] = gg; cs[u] = c_s;
        }
        // G-chunk * ones-columns  (accumulates inferences into cols 0,1)
        v2f ag; ag.x = g[0]; ag.y = g[1];
        acc = __builtin_amdgcn_wmma_f32_16x16x4_f32(
            false, ag, false, b1, (short)0, acc, false, false);
        // S-chunk * action-columns (accumulates colsum·a0 / colsum·a1)
        v2f as2; as2.x = cs[0]; as2.y = cs[1];
        v2f b2;
        b2.x = (col < 2) ? s_act[col * EMB_DIM + d0]     : 0.f;
        b2.y = (col < 2) ? s_act[col * EMB_DIM + d0 + 1] : 0.f;
        acc = __builtin_amdgcn_wmma_f32_16x16x4_f32(
            false, as2, false, b2, (short)0, acc, false, false);
    }

    // D layout: VGPR r -> row r (lanes 0-15) or row 8+r (lanes 16-31); N = col.
    float lpart = 0.f;
    if (col < 2) {
        #pragma unroll
        for (int r = 0; r < 8; ++r) {
            int M = half * 8 + r;
            size_t bg = (size_t)(base_b + wave * BATCH_PER_WAVE + M);
            float infv = acc[r];
            out[bg * 2 + col] = infv;
            float diff = infv - label[bg * 2 + col];
            lpart = fmaf(posw[bg * 2 + col] * diff, diff, lpart);
        }
    }
    // combine lanes {0,16} (inf0 rows) and {1,17} (inf1 rows)
    lpart += __shfl_xor(lpart, 16, 32);
    lpart += __shfl_xor(lpart, 1, 32);
    if (lane == 0) s_red[wave] = lpart;
    __syncthreads();
    if (tid == 0) {
        float s = 0.f;
        #pragma unroll
        for (int w = 0; w < BLOCK / 32; ++w) s += s_red[w];
        blk_loss[blockIdx.x] = s;
    }
}

// Pass 2: deterministic reduction of per-block loss partials -> d_out[2B]
__global__ __launch_bounds__(BLOCK) void fm_reduce_kernel(
    const float* __restrict__ blk_loss, int nblk,
    float* __restrict__ out_scalar, int Bn)
{
    __shared__ float sh[BLOCK];
    const int tid = threadIdx.x;
    float s = 0.f;
    for (int i = tid; i < nblk; i += BLOCK) s += blk_loss[i];
    sh[tid] = s;
    __syncthreads();
    for (int off = BLOCK / 2; off > 0; off >>= 1) {
        if (tid < off) sh[tid] += sh[tid + off];
        __syncthreads();
    }
    if (tid == 0) out_scalar[0] = sh[0] / (float)Bn;
}

extern "C" void kernel_launch(void* const* d_in, const int* in_sizes, int n_in,
                              void* d_out, int out_size, void* d_ws, size_t ws_size,
                              hipStream_t stream) {
    const int*   idx        = (const int*)d_in[0];
    const float* label      = (const float*)d_in[1];
    const float* posw       = (const float*)d_in[2];
    const float* noise      = (const float*)d_in[3];
    const float* emb_mean   = (const float*)d_in[4];
    const float* emb_std    = (const float*)d_in[5];
    const float* fc_w       = (const float*)d_in[6];
    const float* emb_action = (const float*)d_in[7];
    float* out = (float*)d_out;

    const int Bn   = in_sizes[1] / 2;          // label is [B,2]
    const int nblk = Bn / BATCH_PER_BLOCK;     // B = 262144 -> 2048 blocks
    float* blk_loss = (float*)d_ws;

    fm_fused_kernel<<<nblk, BLOCK, 0, stream>>>(
        idx, label, posw, noise, emb_mean, emb_std, fc_w, emb_action,
        out, blk_loss, Bn);
    fm_reduce_kernel<<<1, BLOCK, 0, stream>>>(
        blk_loss, nblk, out + 2 * (size_t)Bn, Bn);
}